// GCN_10780367913710
// MI455X (gfx1250) — compile-verified
//
#include <hip/hip_runtime.h>
#include <hip/hip_bf16.h>

// ---------------------------------------------------------------------------
// GCN forward for MI455X (gfx1250, wave32).
// GEMMs use V_WMMA_F32_16X16X4_F32 (exact fp32 matrix pipe) — one wave per
// 16x16 output tile, K-loop in steps of 4. Edge aggregation is fp32 hardware
// atomics (the bandwidth-bound part; ~50-80us at 23.3 TB/s). Atomics use
// relaxed/agent scope so they lower to single global_atomic_add_f32
// (non-returning, fire-and-forget at L2) instead of a CAS loop.
// ---------------------------------------------------------------------------

typedef __attribute__((ext_vector_type(2))) float v2f;
typedef __attribute__((ext_vector_type(8))) float v8f;

#define N_NODES 50000
#define N_EDGES 800000

// Single hardware f32 atomic add (no return value used -> non-returning op).
__device__ __forceinline__ void atomic_add_f32(float* p, float v) {
    __hip_atomic_fetch_add(p, v, __ATOMIC_RELAXED, __HIP_MEMORY_SCOPE_AGENT);
}

// ---------------- utility kernels ----------------

__global__ void k_fill(float* p, int n, float v) {
    int i = blockIdx.x * blockDim.x + threadIdx.x;
    if (i < n) p[i] = v;
}

__global__ void k_deg(const int* __restrict__ dst, float* deg, int n_edges) {
    int e = blockIdx.x * blockDim.x + threadIdx.x;
    if (e < n_edges) atomic_add_f32(&deg[dst[e]], 1.0f);
}

__global__ void k_rsqrt(float* dis, int n) {
    int i = blockIdx.x * blockDim.x + threadIdx.x;
    if (i < n) {
        float d = dis[i];                 // >= 1 always (self-loop included)
        dis[i] = (d > 0.0f) ? rsqrtf(d) : 0.0f;
    }
}

// ---------------- WMMA f32 GEMM: C[M,N] = A[M,K] @ B[K,N] ----------------
// One wave32 per 16x16 tile of C. K,N compile-time, multiples of 4 / 16.

template <int K, int N>
__global__ void k_gemm_wmma(const float* __restrict__ A, const float* __restrict__ B,
                            float* __restrict__ C, int total_tiles) {
    const int lane = threadIdx.x & 31;
    const int wave = threadIdx.x >> 5;
    const int tile = blockIdx.x * (blockDim.x >> 5) + wave;
    if (tile >= total_tiles) return;          // wave-uniform: EXEC stays all-1s

    constexpr int TILES_N = N / 16;
    const int m_tile = tile / TILES_N;
    const int n_tile = tile % TILES_N;

    const int m  = lane & 15;                 // row within A tile
    const int n  = lane & 15;                 // col within B/C tile
    const int kb = (lane >> 4) << 1;          // lanes 0-15 -> K+0/1, 16-31 -> K+2/3

    const float* Arow = A + (long)(m_tile * 16 + m) * K;
    const float* Bcol = B + n_tile * 16 + n;

    v8f acc = {};
    #pragma unroll
    for (int k = 0; k < K; k += 4) {
        v2f a, b;
        a.x = Arow[k + kb];                   // contiguous pair -> b64 load
        a.y = Arow[k + kb + 1];
        b.x = Bcol[(long)(k + kb) * N];
        b.y = Bcol[(long)(k + kb + 1) * N];
        acc = __builtin_amdgcn_wmma_f32_16x16x4_f32(
            /*neg_a=*/false, a, /*neg_b=*/false, b,
            /*c_mod=*/(short)0, acc, /*reuse_a=*/false, /*reuse_b=*/false);
    }

    // C/D layout: VGPR r, lanes 0-15 -> M=r, lanes 16-31 -> M=r+8, N=lane&15
    float* Crow = C + n_tile * 16 + n;
    const int mbase = m_tile * 16 + ((lane >> 4) << 3);
    #pragma unroll
    for (int r = 0; r < 8; ++r)
        Crow[(long)(mbase + r) * N] = acc[r];
}

// ---------------- edge scatter: agg[dst] += h[src] * dis[src]*dis[dst] -----
// One thread per (edge, 4-feature chunk); float4 loads, f32 atomics.

template <int F>
__global__ void k_scatter(const float* __restrict__ h, const int* __restrict__ src,
                          const int* __restrict__ dst, const float* __restrict__ dis,
                          float* agg, int n_edges) {
    constexpr int VF = F / 4;
    int idx = blockIdx.x * blockDim.x + threadIdx.x;
    int e = idx / VF;
    int f = idx % VF;
    if (e >= n_edges) return;
    int s = src[e], d = dst[e];
    float nm = dis[s] * dis[d];
    const float4* hv = (const float4*)(h + (long)s * F);
    float4 v = hv[f];
    float* o = agg + (long)d * F + f * 4;
    atomic_add_f32(o + 0, v.x * nm);
    atomic_add_f32(o + 1, v.y * nm);
    atomic_add_f32(o + 2, v.z * nm);
    atomic_add_f32(o + 3, v.w * nm);
}

// -------- finalize: out = [relu](agg + h*dis^2 + b); out may alias agg -----

template <int F, bool RELU>
__global__ void k_finalize(const float* __restrict__ h, const float* agg,
                           const float* __restrict__ dis, const float* __restrict__ bias,
                           float* out, int n_nodes) {
    int idx = blockIdx.x * blockDim.x + threadIdx.x;
    int node = idx / F;
    int f = idx % F;
    if (node >= n_nodes) return;
    float sl = dis[node];
    sl *= sl;                                 // self-loop norm = dis^2
    float v = agg[(long)node * F + f] + h[(long)node * F + f] * sl + bias[f];
    if (RELU) v = fmaxf(v, 0.0f);
    out[(long)node * F + f] = v;
}

// -------- head: logits = feat[sel] @ Wl + bl; write lsm then softmax -------

__global__ void k_head(const float* __restrict__ feat, const int* __restrict__ sel,
                       const float* __restrict__ Wl, const float* __restrict__ bl,
                       float* __restrict__ out, int n_sel) {
    int i = blockIdx.x * blockDim.x + threadIdx.x;
    if (i >= n_sel) return;
    const float* x = feat + (long)sel[i] * 32;
    float l0 = bl[0], l1 = bl[1];
    #pragma unroll
    for (int k = 0; k < 32; ++k) {
        float v = x[k];
        l0 += v * Wl[k * 2 + 0];
        l1 += v * Wl[k * 2 + 1];
    }
    float mx = fmaxf(l0, l1);
    float e0 = __expf(l0 - mx), e1 = __expf(l1 - mx);
    float s = e0 + e1;
    float lse = mx + __logf(s);
    out[i * 2 + 0] = l0 - lse;                        // log_softmax
    out[i * 2 + 1] = l1 - lse;
    out[n_sel * 2 + i * 2 + 0] = e0 / s;              // softmax
    out[n_sel * 2 + i * 2 + 1] = e1 / s;
}

// ---------------------------------------------------------------------------

extern "C" void kernel_launch(void* const* d_in, const int* in_sizes, int n_in,
                              void* d_out, int out_size, void* d_ws, size_t ws_size,
                              hipStream_t stream) {
    const float* x   = (const float*)d_in[0];
    const int*   ei  = (const int*)d_in[1];
    const int*   sel = (const int*)d_in[2];
    // d_in[3] = labels (unused by forward outputs)
    const float* W1 = (const float*)d_in[4];
    const float* b1 = (const float*)d_in[5];
    const float* W2 = (const float*)d_in[6];
    const float* b2 = (const float*)d_in[7];
    const float* W3 = (const float*)d_in[8];
    const float* b3 = (const float*)d_in[9];
    const float* Wl = (const float*)d_in[10];
    const float* bl = (const float*)d_in[11];
    float* out = (float*)d_out;

    const int N = N_NODES, E = N_EDGES;
    const int nsel = in_sizes[2];
    const int* src = ei;
    const int* dst = ei + E;

    float* ws  = (float*)d_ws;
    float* dis = ws;                          // 50000 floats
    float* h   = ws + 50048;                  // 50000*128
    float* a   = h + (long)N * 128;           // 50000*128

    const int T = 256;
    auto cdiv = [](long n, long d) { return (int)((n + d - 1) / d); };

    // ---- degree -> dis = rsqrt(deg), self-loop counted via init=1 ----
    k_fill<<<cdiv(N, T), T, 0, stream>>>(dis, N, 1.0f);
    k_deg<<<cdiv(E, T), T, 0, stream>>>(dst, dis, E);
    k_rsqrt<<<cdiv(N, T), T, 0, stream>>>(dis, N);

    // ---- layer 1: h = x @ W1 (50000x128 @ 128x128) ----
    {
        const int tiles = (N / 16) * (128 / 16);
        k_gemm_wmma<128, 128><<<cdiv(tiles, 8), T, 0, stream>>>(x, W1, h, tiles);
        k_fill<<<cdiv((long)N * 128, T), T, 0, stream>>>(a, N * 128, 0.0f);
        k_scatter<128><<<cdiv((long)E * 32, T), T, 0, stream>>>(h, src, dst, dis, a, E);
        k_finalize<128, true><<<cdiv((long)N * 128, T), T, 0, stream>>>(h, a, dis, b1, a, N);
    }

    // ---- layer 2: h = a @ W2 (50000x128 @ 128x32) ----
    {
        const int tiles = (N / 16) * (32 / 16);
        k_gemm_wmma<128, 32><<<cdiv(tiles, 8), T, 0, stream>>>(a, W2, h, tiles);
        k_fill<<<cdiv((long)N * 32, T), T, 0, stream>>>(a, N * 32, 0.0f);
        k_scatter<32><<<cdiv((long)E * 8, T), T, 0, stream>>>(h, src, dst, dis, a, E);
        k_finalize<32, true><<<cdiv((long)N * 32, T), T, 0, stream>>>(h, a, dis, b2, a, N);
    }

    // ---- layer 3: h = a @ W3 (50000x32 @ 32x32), no relu ----
    {
        const int tiles = (N / 16) * (32 / 16);
        k_gemm_wmma<32, 32><<<cdiv(tiles, 8), T, 0, stream>>>(a, W3, h, tiles);
        k_fill<<<cdiv((long)N * 32, T), T, 0, stream>>>(a, N * 32, 0.0f);
        k_scatter<32><<<cdiv((long)E * 8, T), T, 0, stream>>>(h, src, dst, dis, a, E);
        k_finalize<32, false><<<cdiv((long)N * 32, T), T, 0, stream>>>(h, a, dis, b3, a, N);
    }

    // ---- head ----
    k_head<<<cdiv(nsel, T), T, 0, stream>>>(a, sel, Wl, bl, out, nsel);
}